// FusedTensorProductOp4_46497315947105
// MI455X (gfx1250) — compile-verified
//
#include <hip/hip_runtime.h>

typedef __attribute__((ext_vector_type(2))) float v2f;
typedef __attribute__((ext_vector_type(8))) float v8f;
typedef __attribute__((ext_vector_type(4))) unsigned int u32x4;
typedef __attribute__((ext_vector_type(4))) int i32x4;
typedef __attribute__((ext_vector_type(8))) int i32x8;

#if __has_include(<hip/amd_detail/amd_gfx1250_TDM.h>)
#define TDM_6ARG 1
#endif

#define BATCH   1024
#define S_IN    16
#define S_OUT   32
#define IR      3
#define MUL     128
#define NPATH   128
#define DIM_IN  (S_IN * IR * MUL)    /* 6144  */
#define DIM_OUT (S_OUT * IR * MUL)   /* 12288 */
#define SEG     (IR * MUL)           /* 384   */

// Tensor Data Mover: load one contiguous fp32 row (6144 elems = 24 KB) from
// global into LDS at byte offset lds_off. D# per CDNA5 ISA ch.8:
//   group0: [1:0]=count=1, [63:32]=lds_addr, [120:64]=global_addr, [127:126]=type=2
//   group1: [17:16]=data_size=2 (4B), tensor_dim0=6144 @[79:48],
//           tensor_dim1=1 @[111:80], tile_dim0=6144 @[127:112],
//           tile_dim1=1 @[143:128], tensor_dim0_stride=6144 @[207:160]
//   groups 2/3: zero (tensor <= 2D)
__device__ __forceinline__ void tdm_load_row(const float* gsrc, unsigned lds_off)
{
    const unsigned long long ga = (unsigned long long)(uintptr_t)gsrc;
    u32x4 g0;
    g0.x = 1u;                                                  // count=1 (valid)
    g0.y = lds_off;                                             // lds_addr (bytes)
    g0.z = (unsigned)(ga & 0xFFFFFFFFull);                      // global_addr lo
    g0.w = (unsigned)((ga >> 32) & 0x01FFFFFFull) | (2u << 30); // addr hi | type=2

    i32x8 g1;
    g1[0] = 0x20000;                                // data_size=2 -> 4 bytes
    g1[1] = (int)((DIM_IN & 0xFFFF) << 16);         // tensor_dim0 lo16
    g1[2] = (int)((DIM_IN >> 16) | (1u << 16));     // tensor_dim0 hi16 | tensor_dim1 lo16
    g1[3] = (int)((unsigned)DIM_IN << 16);          // tensor_dim1 hi16 | tile_dim0
    g1[4] = 1;                                      // tile_dim1=1, tile_dim2=0
    g1[5] = DIM_IN;                                 // tensor_dim0_stride lo32
    g1[6] = 0;                                      // stride0 hi16 | stride1 lo16
    g1[7] = 0;                                      // stride1 hi32

    i32x4 gz = {0, 0, 0, 0};
#ifdef TDM_6ARG
    i32x8 gz8 = {0, 0, 0, 0, 0, 0, 0, 0};
    __builtin_amdgcn_tensor_load_to_lds(g0, g1, gz, gz, gz8, 0);
#else
    __builtin_amdgcn_tensor_load_to_lds(g0, g1, gz, gz, 0);
#endif
}

// One workgroup per batch element. 8 wave32s; each wave owns 16 paths.
// Per (path, n-tile): build T(27x16) per-lane in registers, 7 x
// v_wmma_f32_16x16x4_f32 against the per-path CG matrix (3x27 in a 16-row
// tile), LDS-atomic-add rows 0..2 of D into the output accumulator.
__global__ __launch_bounds__(256)
void stp_wmma_kernel(const float* __restrict__ x0,
                     const float* __restrict__ x1,
                     const float* __restrict__ x2,
                     const float* __restrict__ coeffs,
                     const int*   __restrict__ idx0,
                     const int*   __restrict__ idx1,
                     const int*   __restrict__ idx2,
                     const int*   __restrict__ idx3,
                     float*       __restrict__ out)
{
    extern __shared__ float smem[];
    float* lx0  = smem;                  // 6144 floats @ LDS byte 0
    float* lx1  = smem + DIM_IN;         // 6144 @ byte 24576
    float* lx2  = smem + 2 * DIM_IN;     // 6144 @ byte 49152
    float* lout = smem + 3 * DIM_IN;     // 12288

    const int b    = blockIdx.x;
    const int tid  = threadIdx.x;
    const int lane = tid & 31;
    const int wave = tid >> 5;           // 0..7
    const int row  = lane & 15;          // M-row / N-col position
    const int hi   = lane >> 4;          // 0: K pair {0,1}, 1: K pair {2,3}

    // ---- stage inputs to LDS via TDM (wave 0), zero accumulator (all) ----
    if (tid < 32) {
        tdm_load_row(x0 + (size_t)b * DIM_IN, 0u);
        tdm_load_row(x1 + (size_t)b * DIM_IN, (unsigned)(DIM_IN * 4));
        tdm_load_row(x2 + (size_t)b * DIM_IN, (unsigned)(2 * DIM_IN * 4));
    }
    {
        float4  z  = make_float4(0.f, 0.f, 0.f, 0.f);
        float4* so = (float4*)lout;
        for (int t = tid; t < DIM_OUT / 4; t += 256) so[t] = z;
    }
    __builtin_amdgcn_s_wait_tensorcnt(0);   // TDM tiles landed in LDS
    __syncthreads();

    for (int p = wave; p < NPATH; p += 8) {
        const int s0i = idx0[p];
        const int s1i = idx1[p];
        const int s2i = idx2[p];
        const int s3i = idx3[p];
        const float* Cp = coeffs + p * 81;   // C[p, i,j,k, l] -> Cp[ijk*3 + l]

        if (p + 8 < NPATH)                   // global_prefetch_b8
            __builtin_prefetch(coeffs + (p + 8) * 81, 0, 3);

        // A fragments: A[M=l][K=ijk], rows 3..15 zero, K padded 27 -> 28.
        // ISA 16x4 f32 A layout: lanes 0-15 M=0..15; vgpr0 = K, K+2(hi);
        // vgpr1 = K+1, K+3(hi).
        v2f a[7];
        #pragma unroll
        for (int kk = 0; kk < 7; ++kk) {
            const int kb = kk * 4 + hi * 2;  // 0..26
            float ax = 0.f, ay = 0.f;
            if (row < IR) {
                ax = Cp[kb * 3 + row];
                if (kb + 1 < 27) ay = Cp[(kb + 1) * 3 + row];
            }
            a[kk].x = ax;
            a[kk].y = ay;
        }

        #pragma unroll
        for (int nt = 0; nt < 8; ++nt) {
            const int u = nt * 16 + row;

            float r0[IR], r1[IR], r2[IR];
            #pragma unroll
            for (int i = 0; i < IR; ++i) {
                r0[i] = lx0[s0i * SEG + i * MUL + u];
                r1[i] = lx1[s1i * SEG + i * MUL + u];
                r2[i] = lx2[s2i * SEG + i * MUL + u];
            }

            // T[ijk] = g0[i]*g1[j]*g2[k] for this lane's u (36 v_mul)
            float T[28];
            #pragma unroll
            for (int i = 0; i < IR; ++i) {
                #pragma unroll
                for (int j = 0; j < IR; ++j) {
                    const float r01 = r0[i] * r1[j];
                    #pragma unroll
                    for (int k = 0; k < IR; ++k)
                        T[(i * 3 + j) * 3 + k] = r01 * r2[k];
                }
            }
            T[27] = 0.f;   // K padding

            v8f acc = {0.f, 0.f, 0.f, 0.f, 0.f, 0.f, 0.f, 0.f};
            #pragma unroll
            for (int kk = 0; kk < 7; ++kk) {
                // B 4x16 layout: lanes 0-15 N-col, vgpr0 = row K (lo) /
                // K+2 (hi lanes); vgpr1 = K+1 / K+3. Compile-time indices
                // + v_cndmask, no dynamic register indexing.
                v2f bf;
                bf.x = hi ? T[kk * 4 + 2] : T[kk * 4 + 0];
                bf.y = hi ? T[kk * 4 + 3] : T[kk * 4 + 1];
                acc = __builtin_amdgcn_wmma_f32_16x16x4_f32(
                    false, a[kk], false, bf, (short)0, acc, false, false);
            }

            // D rows 0..2 live in acc[0..2] of lanes 0-15 (col N = lane).
            if (lane < 16) {
                float* dst = lout + (s3i * IR) * MUL + nt * 16 + lane;
                atomicAdd(dst + 0 * MUL, acc[0]);
                atomicAdd(dst + 1 * MUL, acc[1]);
                atomicAdd(dst + 2 * MUL, acc[2]);
            }
        }
    }
    __syncthreads();

    // ---- drain accumulator to global (single full-row write) ----
    {
        float4*       go = (float4*)(out + (size_t)b * DIM_OUT);
        const float4* so = (const float4*)lout;
        for (int t = tid; t < DIM_OUT / 4; t += 256) go[t] = so[t];
    }
}

extern "C" void kernel_launch(void* const* d_in, const int* in_sizes, int n_in,
                              void* d_out, int out_size, void* d_ws, size_t ws_size,
                              hipStream_t stream) {
    (void)in_sizes; (void)n_in; (void)out_size; (void)d_ws; (void)ws_size;
    const float* x0     = (const float*)d_in[0];
    const float* x1     = (const float*)d_in[1];
    const float* x2     = (const float*)d_in[2];
    const float* coeffs = (const float*)d_in[3];
    const int*   idx0   = (const int*)d_in[4];
    const int*   idx1   = (const int*)d_in[5];
    const int*   idx2   = (const int*)d_in[6];
    const int*   idx3   = (const int*)d_in[7];
    float*       out    = (float*)d_out;

    const size_t shmem = (size_t)(3 * DIM_IN + DIM_OUT) * sizeof(float); // 120 KB
    stp_wmma_kernel<<<dim3(BATCH), dim3(256), shmem, stream>>>(
        x0, x1, x2, coeffs, idx0, idx1, idx2, idx3, out);
}